// PRMLayer_58042188038397
// MI455X (gfx1250) — compile-verified
//
#include <hip/hip_runtime.h>
#include <math.h>

#define GROUPS 52
#define BATCH  16
#define HDIM   256
#define WDIM   256
#define HW     (HDIM * WDIM)      // 65536 elements per (b,g) row
#define NT     1024               // threads per block = 32 wave32 waves
#define NW     (NT / 32)
#define ITERS  (HW / (NT * 4))    // 16 float4 iterations per thread

// ---- dynamic LDS layout (in floats) ----
#define OFF_X   0                 // [0, 65536)   : x row
#define OFF_S1  HW                // 32 partials
#define OFF_S2  (HW + 32)         // 32 partials
#define OFF_SI  (HW + 64)         // 32 int partials (argmax index)
#define OFF_BC  (HW + 96)         // 16 broadcast scalars
#define OFF_R   (HW + 112)        // 256-entry row table: 0.5*A*gpdf_row + Cc
#define OFF_C   (HW + 368)        // 256-entry col table: 0.5*A*gpdf_col
#define SMEM_FLOATS (HW + 624)
#define SMEM_BYTES  (SMEM_FLOATS * 4)   // 264,640 B  (< 320 KB WGP LDS)

#ifndef __has_builtin
#define __has_builtin(x) 0
#endif

typedef int   v4i __attribute__((ext_vector_type(4)));
typedef float v4f __attribute__((ext_vector_type(4)));
typedef __attribute__((address_space(1))) v4i* gbl_v4i_ptr;
typedef __attribute__((address_space(3))) v4i* lds_v4i_ptr;

// Asynchronous global -> LDS copy of 16 bytes (CDNA5 ASYNCcnt path).
__device__ __forceinline__ void async_copy_b128(const float* gsrc, float* ldst) {
#if __has_builtin(__builtin_amdgcn_global_load_async_to_lds_b128)
  __builtin_amdgcn_global_load_async_to_lds_b128(
      (gbl_v4i_ptr)(size_t)gsrc,
      (lds_v4i_ptr)(size_t)(unsigned)(size_t)ldst,  // low 32 bits = LDS byte address
      /*imm offset=*/0, /*cpol=*/0);
#else
  unsigned lofs = (unsigned)(size_t)ldst;
  asm volatile("global_load_async_to_lds_b128 %0, %1, off"
               :: "v"(lofs), "v"(gsrc) : "memory");
#endif
}

__device__ __forceinline__ void wait_async0() {
#if __has_builtin(__builtin_amdgcn_s_wait_asynccnt)
  __builtin_amdgcn_s_wait_asynccnt(0);
#else
  asm volatile("s_wait_asynccnt 0" ::: "memory");
#endif
}

__global__ void __launch_bounds__(NT)
prm_layer_kernel(const float* __restrict__ x,
                 const float* __restrict__ weight,
                 const float* __restrict__ bias,
                 const float* __restrict__ one_,
                 const float* __restrict__ zero_,
                 const float* __restrict__ theta,
                 const float* __restrict__ scale,
                 float* __restrict__ out) {
  extern __shared__ float smem[];
  const int tid  = threadIdx.x;
  const int lane = tid & 31;
  const int wave = tid >> 5;
  const int bg   = blockIdx.x;          // (b,g) pair, 0..831
  const int g    = bg % GROUPS;
  const size_t base = (size_t)bg * HW;
  const float* xrow = x + base;

  float* S1 = smem + OFF_S1;
  float* S2 = smem + OFF_S2;
  int*   SI = (int*)(smem + OFF_SI);
  float* BC = smem + OFF_BC;
  float* Rt = smem + OFF_R;
  float* Ct = smem + OFF_C;

  // ---------------- Phase 0: async DMA the 256KB row into LDS ----------------
#pragma unroll
  for (int k = 0; k < ITERS; ++k) {
    const int e4 = tid + k * NT;               // float4 index
    async_copy_b128(xrow + 4 * e4, smem + 4 * e4);
  }
  wait_async0();        // drain this wave's ASYNCcnt
  __syncthreads();      // ensure all waves' DMA landed before any LDS read

  const float4* xl = (const float4*)smem;

  // ---------------- Phase 1: argmax (first occurrence) + sum ----------------
  float vmax = -INFINITY;
  int   vidx = 0;
  float fsum = 0.0f;
#pragma unroll
  for (int k = 0; k < ITERS; ++k) {
    const int e4 = tid + k * NT;
    const float4 v = xl[e4];
    const int e = 4 * e4;
    fsum += v.x + v.y + v.z + v.w;
    if (v.x > vmax) { vmax = v.x; vidx = e; }
    if (v.y > vmax) { vmax = v.y; vidx = e + 1; }
    if (v.z > vmax) { vmax = v.z; vidx = e + 2; }
    if (v.w > vmax) { vmax = v.w; vidx = e + 3; }
  }
#pragma unroll
  for (int o = 16; o > 0; o >>= 1) {
    const float ov = __shfl_xor(vmax, o, 32);
    const int   oi = __shfl_xor(vidx, o, 32);
    const float os = __shfl_xor(fsum, o, 32);
    fsum += os;
    if (ov > vmax || (ov == vmax && oi < vidx)) { vmax = ov; vidx = oi; }
  }
  if (lane == 0) { S1[wave] = vmax; SI[wave] = vidx; S2[wave] = fsum; }
  __syncthreads();
  if (wave == 0) {
    float v  = S1[lane];
    int   i  = SI[lane];
    float ps = S2[lane];
#pragma unroll
    for (int o = 16; o > 0; o >>= 1) {
      const float ov = __shfl_xor(v, o, 32);
      const int   oi = __shfl_xor(i, o, 32);
      const float os = __shfl_xor(ps, o, 32);
      ps += os;
      if (ov > v || (ov == v && oi < i)) { v = ov; i = oi; }
    }
    if (lane == 0) {
      BC[0] = v;                         // qval
      ((int*)BC)[1] = i;                 // argmax linear index
      BC[2] = ps * (1.0f / HW);          // gap (spatial mean)
    }
  }
  __syncthreads();

  // -------- Phase 1b: pre-scaled factorized Gaussian tables --------
  // A*dist + Cc == (0.5A*gpdf_r[row] + Cc) + (0.5A*gpdf_c[col])
  const float qval = BC[0];
  const float gap  = BC[2];
  const float A     = qval * zero_[g];        // qval * zero[g]
  const float Cc    = gap * one_[g];          // gap  * one[g]
  const float halfA = 0.5f * A;
  const float sigma     = scale[0];
  const float inv_sigma = 1.0f / sigma;
  const float log_norm  = -logf(sigma) - 0.91893853320467274178f; // -0.5*log(2*pi)
  const float th0 = theta[0];
  const float th1 = theta[1];
  const int   qidx = ((int*)BC)[1];
  const float qrow = (float)(qidx >> 8);     // / WDIM
  const float qcol = (float)(qidx & 255);    // % WDIM
  if (tid < HDIM) {
    const float zr = fabsf((float)tid - qrow) * th0 * inv_sigma;
    const float zc = fabsf((float)tid - qcol) * th1 * inv_sigma;
    Rt[tid] = fmaf(halfA, expf(-0.5f * zr * zr + log_norm), Cc);
    Ct[tid] = halfA * expf(-0.5f * zc * zc + log_norm);
  }
  __syncthreads();

  // ---------------- Phase 2: mean / std of sim = x*(Rt[row] + Ct[col]) ----------------
  float s1 = 0.0f, s2 = 0.0f;
#pragma unroll
  for (int k = 0; k < ITERS; ++k) {
    const int e4 = tid + k * NT;
    const float4 v = xl[e4];
    const int e = 4 * e4;
    const float r  = Rt[e >> 8];         // same row for all 4 lanes of the float4
    const int  c0  = e & 255;
    const float m0 = v.x * (r + Ct[c0]);
    const float m1 = v.y * (r + Ct[c0 + 1]);
    const float m2 = v.z * (r + Ct[c0 + 2]);
    const float m3 = v.w * (r + Ct[c0 + 3]);
    s1 += m0 + m1 + m2 + m3;
    s2 += m0 * m0 + m1 * m1 + m2 * m2 + m3 * m3;
  }
#pragma unroll
  for (int o = 16; o > 0; o >>= 1) {
    s1 += __shfl_xor(s1, o, 32);
    s2 += __shfl_xor(s2, o, 32);
  }
  if (lane == 0) { S1[wave] = s1; S2[wave] = s2; }
  __syncthreads();
  if (wave == 0) {
    float p1 = S1[lane], p2 = S2[lane];
#pragma unroll
    for (int o = 16; o > 0; o >>= 1) {
      p1 += __shfl_xor(p1, o, 32);
      p2 += __shfl_xor(p2, o, 32);
    }
    if (lane == 0) {
      const float mean = p1 * (1.0f / HW);
      float var = (p2 - (float)HW * mean * mean) * (1.0f / (HW - 1));
      var = var > 0.0f ? var : 0.0f;
      BC[4] = mean;
      BC[5] = 1.0f / (sqrtf(var) + 1e-5f);   // 1 / (std + 1e-5)
    }
  }
  __syncthreads();

  // ------------- Phase 3: out = x * sigmoid((sim-mean)*istd * w + b) -------------
  const float mean = BC[4];
  const float istd = BC[5];
  const float wi  = weight[g] * istd;        // fold istd into weight
  const float bgc = fmaf(-mean * istd, weight[g], bias[g]);  // ctx = sim*wi + bgc
  v4f* orow = (v4f*)(out + base);
#pragma unroll
  for (int k = 0; k < ITERS; ++k) {
    const int e4 = tid + k * NT;
    const float4 v = xl[e4];
    const int e = 4 * e4;
    const float r  = Rt[e >> 8];
    const int  c0  = e & 255;
    const float m0 = v.x * (r + Ct[c0]);
    const float m1 = v.y * (r + Ct[c0 + 1]);
    const float m2 = v.z * (r + Ct[c0 + 2]);
    const float m3 = v.w * (r + Ct[c0 + 3]);
    const float c0x = fmaf(m0, wi, bgc);
    const float c1x = fmaf(m1, wi, bgc);
    const float c2x = fmaf(m2, wi, bgc);
    const float c3x = fmaf(m3, wi, bgc);
    v4f res;
    res.x = v.x / (1.0f + expf(-c0x));
    res.y = v.y / (1.0f + expf(-c1x));
    res.z = v.z / (1.0f + expf(-c2x));
    res.w = v.w / (1.0f + expf(-c3x));
    __builtin_nontemporal_store(res, orow + e4);   // write-once stream, keep L2 clean
  }
}

extern "C" void kernel_launch(void* const* d_in, const int* in_sizes, int n_in,
                              void* d_out, int out_size, void* d_ws, size_t ws_size,
                              hipStream_t stream) {
  const float* x      = (const float*)d_in[0];
  const float* weight = (const float*)d_in[1];
  const float* bias   = (const float*)d_in[2];
  const float* one_   = (const float*)d_in[3];
  const float* zero_  = (const float*)d_in[4];
  const float* theta  = (const float*)d_in[5];
  const float* scale  = (const float*)d_in[6];
  float* out = (float*)d_out;

  dim3 grid(BATCH * GROUPS);   // 832 blocks, one (b,g) row each
  prm_layer_kernel<<<grid, NT, SMEM_BYTES, stream>>>(
      x, weight, bias, one_, zero_, theta, scale, out);
}